// PlayGAT_1090921693627
// MI455X (gfx1250) — compile-verified
//
#include <hip/hip_runtime.h>
#include <math.h>

typedef __attribute__((ext_vector_type(2))) float v2f;
typedef __attribute__((ext_vector_type(8))) float v8f;

#define NEG_SLOPE 0.2f
#define EPS_DIV   1e-16f

// ---------------------------------------------------------------------------
// helpers
// ---------------------------------------------------------------------------
static __device__ inline v8f wmma_f32_k4(v2f a, v2f b, v8f c) {
  // D = A(16x4 f32) * B(4x16 f32) + C(16x16 f32)
  return __builtin_amdgcn_wmma_f32_16x16x4_f32(
      /*neg_a=*/false, a, /*neg_b=*/false, b,
      /*c_mod=*/(short)0, c, /*reuse_a=*/false, /*reuse_b=*/false);
}

static __device__ inline void atomicMaxF(float* addr, float v) {
  // IEEE-754 bit trick: positives via signed-int max, negatives via unsigned min.
  if (v >= 0.0f) atomicMax((int*)addr, __float_as_int(v));
  else           atomicMin((unsigned int*)addr, (unsigned int)__float_as_uint(v));
}

__global__ void fill_f32(float* p, float v, int n) {
  int i = blockIdx.x * blockDim.x + threadIdx.x;
  if (i < n) p[i] = v;
}

// ---------------------------------------------------------------------------
// weight staging: transpose+pad so WMMA B fragments are contiguous float2 pairs
//   W1t[128][8] : W1t[n*8+k]   = (k<7) ? W1[k*128+n] : 0
//   W2t[32][128]: W2t[n*128+k] = W2[k*32+n]
// ---------------------------------------------------------------------------
__global__ void prep_weights(const float* __restrict__ W1, const float* __restrict__ W2,
                             float* __restrict__ W1t, float* __restrict__ W2t) {
  int i = blockIdx.x * blockDim.x + threadIdx.x;
  if (i < 1024) {
    int n = i >> 3, k = i & 7;
    W1t[i] = (k < 7) ? W1[k * 128 + n] : 0.f;
  } else if (i < 1024 + 4096) {
    int j = i - 1024;
    int n = j >> 7, k = j & 127;
    W2t[j] = W2[k * 32 + n];
  }
}

// xpad[N][8]: zero-padded K so A fragments load unconditionally
__global__ void pad_x(const float* __restrict__ x, float* __restrict__ xpad, int total) {
  int i = blockIdx.x * blockDim.x + threadIdx.x;
  if (i >= total) return;
  int n = i >> 3, k = i & 7;
  xpad[i] = (k < 7) ? x[n * 7 + k] : 0.f;
}

// ---------------------------------------------------------------------------
// fold att_e into edge_attr space:  Me1[7][4], Me2[7]
// ---------------------------------------------------------------------------
__global__ void fold_atte(const float* __restrict__ We1, const float* __restrict__ att_e1,
                          const float* __restrict__ We2, const float* __restrict__ att_e2,
                          float* __restrict__ Me1, float* __restrict__ Me2) {
  int t = threadIdx.x;
  if (t < 28) {                         // Me1[d][h] = sum_c We1[d][h*32+c]*att_e1[h][c]
    int d = t >> 2, h = t & 3;
    float s = 0.f;
    for (int c = 0; c < 32; ++c) s += We1[d * 128 + h * 32 + c] * att_e1[h * 32 + c];
    Me1[t] = s;
  } else if (t >= 32 && t < 39) {       // Me2[d] = sum_c We2[d][c]*att_e2[c]
    int d = t - 32;
    float s = 0.f;
    for (int c = 0; c < 32; ++c) s += We2[d * 32 + c] * att_e2[c];
    Me2[d] = s;
  }
}

// ---------------------------------------------------------------------------
// GEMM 1:  h1[N,128] = xpad[N,8] @ W1pad[8,128]   (2 WMMA k-steps, 8 col tiles)
// one wave per 16-row tile; unconditional b64 loads, full-EXEC (WMMA rule).
// N % 16 == 0.
// ---------------------------------------------------------------------------
__global__ __launch_bounds__(32)
void gemm_x_w1(const float* __restrict__ xpad, const float* __restrict__ W1t,
               float* __restrict__ h1) {
  const int lane  = threadIdx.x & 31;
  const int ln    = lane & 15;
  const int khalf = lane >> 4;
  const int m0    = blockIdx.x * 16;
  const int m     = m0 + ln;

  v8f acc[8];
  const v8f vzero = {0.f, 0.f, 0.f, 0.f, 0.f, 0.f, 0.f, 0.f};
#pragma unroll
  for (int t = 0; t < 8; ++t) acc[t] = vzero;

#pragma unroll
  for (int kk = 0; kk < 8; kk += 4) {
    const int kb = kk + 2 * khalf;
    const v2f a = *(const v2f*)&xpad[m * 8 + kb];
#pragma unroll
    for (int t = 0; t < 8; ++t) {
      const int n = t * 16 + ln;
      const v2f b = *(const v2f*)&W1t[n * 8 + kb];
      acc[t] = wmma_f32_k4(a, b, acc[t]);
    }
  }
#pragma unroll
  for (int t = 0; t < 8; ++t) {
    const int n = t * 16 + ln;
#pragma unroll
    for (int r = 0; r < 8; ++r)
      h1[(m0 + r + 8 * khalf) * 128 + n] = acc[t][r];
  }
}

// ---------------------------------------------------------------------------
// GEMM 2:  h2[N,32] = hbn[N,128] @ W2[128,32]   (32 WMMA k-steps, 2 col tiles)
// ---------------------------------------------------------------------------
__global__ __launch_bounds__(32)
void gemm_h_w2(const float* __restrict__ hbn, const float* __restrict__ W2t,
               float* __restrict__ h2) {
  const int lane  = threadIdx.x & 31;
  const int ln    = lane & 15;
  const int khalf = lane >> 4;
  const int m0    = blockIdx.x * 16;
  const int m     = m0 + ln;

  const v8f vzero = {0.f, 0.f, 0.f, 0.f, 0.f, 0.f, 0.f, 0.f};
  v8f acc0 = vzero, acc1 = vzero;

#pragma unroll 4
  for (int kk = 0; kk < 128; kk += 4) {
    const int kb = kk + 2 * khalf;
    const v2f a  = *(const v2f*)&hbn[m * 128 + kb];
    const v2f b0 = *(const v2f*)&W2t[ln * 128 + kb];
    const v2f b1 = *(const v2f*)&W2t[(16 + ln) * 128 + kb];
    acc0 = wmma_f32_k4(a, b0, acc0);
    acc1 = wmma_f32_k4(a, b1, acc1);
  }
#pragma unroll
  for (int r = 0; r < 8; ++r) {
    const int mo = m0 + r + 8 * khalf;
    h2[mo * 32 + ln]      = acc0[r];
    h2[mo * 32 + 16 + ln] = acc1[r];
  }
}

// ---------------------------------------------------------------------------
// per-node attention dots
// ---------------------------------------------------------------------------
__global__ void node_att1(const float* __restrict__ h1,
                          const float* __restrict__ att_s, const float* __restrict__ att_d,
                          float* __restrict__ a_s, float* __restrict__ a_d, int Nn) {
  int i = blockIdx.x * blockDim.x + threadIdx.x;   // (n,h)
  if (i >= Nn * 4) return;
  int n = i >> 2, h = i & 3;
  float ss = 0.f, sd = 0.f;
  for (int c = 0; c < 32; ++c) {
    float v = h1[n * 128 + h * 32 + c];
    ss += v * att_s[h * 32 + c];
    sd += v * att_d[h * 32 + c];
  }
  a_s[i] = ss;
  a_d[i] = sd;
}

__global__ void node_att2(const float* __restrict__ h2,
                          const float* __restrict__ att_s, const float* __restrict__ att_d,
                          float* __restrict__ a_s, float* __restrict__ a_d, int Nn) {
  int n = blockIdx.x * blockDim.x + threadIdx.x;
  if (n >= Nn) return;
  float ss = 0.f, sd = 0.f;
  for (int c = 0; c < 32; ++c) {
    float v = h2[n * 32 + c];
    ss += v * att_s[c];
    sd += v * att_d[c];
  }
  a_s[n] = ss;
  a_d[n] = sd;
}

// ---------------------------------------------------------------------------
// edge passes, layer 1 (H=4)
// ---------------------------------------------------------------------------
__global__ void edge_alpha1(const int* __restrict__ src, const int* __restrict__ dst,
                            const float* __restrict__ eattr, const float* __restrict__ Me1,
                            const float* __restrict__ a_s, const float* __restrict__ a_d,
                            float* __restrict__ alpha, float* __restrict__ mbuf, int Ee) {
  int e = blockIdx.x * blockDim.x + threadIdx.x;
  if (e >= Ee) return;
  int s = src[e], d = dst[e];
  float ea[7];
  for (int j = 0; j < 7; ++j) ea[j] = eattr[e * 7 + j];
  for (int h = 0; h < 4; ++h) {
    float ae = 0.f;
    for (int j = 0; j < 7; ++j) ae += ea[j] * Me1[j * 4 + h];
    float a = a_s[s * 4 + h] + a_d[d * 4 + h] + ae;
    a = (a > 0.f) ? a : NEG_SLOPE * a;
    alpha[e * 4 + h] = a;
    atomicMaxF(&mbuf[d * 4 + h], a);
  }
}

__global__ void edge_exp1(const int* __restrict__ dst, const float* __restrict__ mbuf,
                          float* __restrict__ alpha, float* __restrict__ sbuf, int Ee) {
  int e = blockIdx.x * blockDim.x + threadIdx.x;
  if (e >= Ee) return;
  int d = dst[e];
  for (int h = 0; h < 4; ++h) {
    float ex = __expf(alpha[e * 4 + h] - mbuf[d * 4 + h]);
    alpha[e * 4 + h] = ex;
    atomicAdd(&sbuf[d * 4 + h], ex);
  }
}

__global__ void edge_msg1(const int* __restrict__ src, const int* __restrict__ dst,
                          const float* __restrict__ h1, const float* __restrict__ alpha,
                          const float* __restrict__ sbuf, float* __restrict__ out1,
                          int total) {
  int i = blockIdx.x * blockDim.x + threadIdx.x;   // (e, 32 chunks of 4 ch)
  if (i >= total) return;
  int e  = i >> 5;
  int g  = i & 31;
  int h  = g >> 3;
  int c4 = (g & 7) * 4;
  int s = src[e], d = dst[e];
  float al = alpha[e * 4 + h] / (sbuf[d * 4 + h] + EPS_DIV);
  const float4 hv = *(const float4*)&h1[s * 128 + h * 32 + c4];
  float* o = &out1[d * 128 + h * 32 + c4];
  atomicAdd(o + 0, hv.x * al);
  atomicAdd(o + 1, hv.y * al);
  atomicAdd(o + 2, hv.z * al);
  atomicAdd(o + 3, hv.w * al);
}

// ---------------------------------------------------------------------------
// bias + batchnorm(eval) + ELU  ->  hbn
// ---------------------------------------------------------------------------
__global__ void bn_elu(const float* __restrict__ out1, const float* __restrict__ b1,
                       const float* __restrict__ g, const float* __restrict__ bb,
                       const float* __restrict__ mm, const float* __restrict__ vv,
                       float* __restrict__ hbn, int total) {
  int i = blockIdx.x * blockDim.x + threadIdx.x;
  if (i >= total) return;
  int c = i & 127;
  float y = out1[i] + b1[c];
  y = (y - mm[c]) * rsqrtf(vv[c] + 1e-5f) * g[c] + bb[c];
  hbn[i] = (y > 0.f) ? y : (__expf(y) - 1.f);
}

// ---------------------------------------------------------------------------
// edge passes, layer 2 (H=1)
// ---------------------------------------------------------------------------
__global__ void edge_alpha2(const int* __restrict__ src, const int* __restrict__ dst,
                            const float* __restrict__ eattr, const float* __restrict__ Me2,
                            const float* __restrict__ a_s, const float* __restrict__ a_d,
                            float* __restrict__ alpha, float* __restrict__ mbuf, int Ee) {
  int e = blockIdx.x * blockDim.x + threadIdx.x;
  if (e >= Ee) return;
  int s = src[e], d = dst[e];
  float ae = 0.f;
  for (int j = 0; j < 7; ++j) ae += eattr[e * 7 + j] * Me2[j];
  float a = a_s[s] + a_d[d] + ae;
  a = (a > 0.f) ? a : NEG_SLOPE * a;
  alpha[e] = a;
  atomicMaxF(&mbuf[d], a);
}

__global__ void edge_exp2(const int* __restrict__ dst, const float* __restrict__ mbuf,
                          float* __restrict__ alpha, float* __restrict__ sbuf, int Ee) {
  int e = blockIdx.x * blockDim.x + threadIdx.x;
  if (e >= Ee) return;
  int d = dst[e];
  float ex = __expf(alpha[e] - mbuf[d]);
  alpha[e] = ex;
  atomicAdd(&sbuf[d], ex);
}

__global__ void edge_msg2(const int* __restrict__ src, const int* __restrict__ dst,
                          const float* __restrict__ h2, const float* __restrict__ alpha,
                          const float* __restrict__ sbuf, float* __restrict__ out2,
                          int total) {
  int i = blockIdx.x * blockDim.x + threadIdx.x;   // (e, 8 chunks of 4 ch)
  if (i >= total) return;
  int e  = i >> 3;
  int c4 = (i & 7) * 4;
  int s = src[e], d = dst[e];
  float al = alpha[e] / (sbuf[d] + EPS_DIV);
  const float4 hv = *(const float4*)&h2[s * 32 + c4];
  float* o = &out2[d * 32 + c4];
  atomicAdd(o + 0, hv.x * al);
  atomicAdd(o + 1, hv.y * al);
  atomicAdd(o + 2, hv.z * al);
  atomicAdd(o + 3, hv.w * al);
}

// ---------------------------------------------------------------------------
// bias + ELU + global mean-pool accumulation
// ---------------------------------------------------------------------------
__global__ void elu_pool(const float* __restrict__ out2, const float* __restrict__ b2,
                         const int* __restrict__ batch, float* __restrict__ pooled,
                         float* __restrict__ cnt, int Nn) {
  int i = blockIdx.x * blockDim.x + threadIdx.x;
  if (i >= Nn * 32) return;
  int n = i >> 5, c = i & 31;
  float y = out2[i] + b2[c];
  y = (y > 0.f) ? y : (__expf(y) - 1.f);
  int gi = batch[n];
  atomicAdd(&pooled[gi * 32 + c], y);
  if (c == 0) atomicAdd(&cnt[gi], 1.0f);
}

__global__ void head(const float* __restrict__ pooled, const float* __restrict__ cnt,
                     const float* __restrict__ Wh, const float* __restrict__ bh,
                     float* __restrict__ out, int Gg) {
  int g = blockIdx.x * blockDim.x + threadIdx.x;
  if (g >= Gg) return;
  float denom = fmaxf(cnt[g], 1.0f);
  float s = 0.f;
  for (int c = 0; c < 32; ++c) s += (pooled[g * 32 + c] / denom) * Wh[c];
  out[g] = s + bh[0];
}

// ---------------------------------------------------------------------------
// launch
// ---------------------------------------------------------------------------
extern "C" void kernel_launch(void* const* d_in, const int* in_sizes, int n_in,
                              void* d_out, int out_size, void* d_ws, size_t ws_size,
                              hipStream_t stream) {
  const float* x      = (const float*)d_in[0];
  const int*   eidx   = (const int*)d_in[1];
  const float* eattr  = (const float*)d_in[2];
  const int*   batch  = (const int*)d_in[3];
  const float* W1     = (const float*)d_in[4];
  const float* att_s1 = (const float*)d_in[5];
  const float* att_d1 = (const float*)d_in[6];
  const float* We1    = (const float*)d_in[7];
  const float* att_e1 = (const float*)d_in[8];
  const float* b1     = (const float*)d_in[9];
  const float* bn_g   = (const float*)d_in[10];
  const float* bn_b   = (const float*)d_in[11];
  const float* bn_m   = (const float*)d_in[12];
  const float* bn_v   = (const float*)d_in[13];
  const float* W2     = (const float*)d_in[14];
  const float* att_s2 = (const float*)d_in[15];
  const float* att_d2 = (const float*)d_in[16];
  const float* We2    = (const float*)d_in[17];
  const float* att_e2 = (const float*)d_in[18];
  const float* b2     = (const float*)d_in[19];
  const float* Wh     = (const float*)d_in[20];
  const float* bh     = (const float*)d_in[21];
  float* out = (float*)d_out;

  const int N = in_sizes[0] / 7;     // 50000 (multiple of 16)
  const int E = in_sizes[2] / 7;     // 800000
  const int G = out_size;            // 500
  const int* src = eidx;             // edge_index[0]
  const int* dst = eidx + E;         // edge_index[1]

  // workspace carve-up (floats)
  float* w = (float*)d_ws;
  size_t off = 0;
  float* h1     = w + off; off += (size_t)N * 128;   // later reused as hbn
  float* out1   = w + off; off += (size_t)N * 128;
  float* xpad   = w + off; off += (size_t)N * 8;
  float* as1    = w + off; off += (size_t)N * 4;
  float* ad1    = w + off; off += (size_t)N * 4;
  float* m1     = w + off; off += (size_t)N * 4;
  float* s1     = w + off; off += (size_t)N * 4;
  float* alpha1 = w + off; off += (size_t)E * 4;
  float* h2     = w + off; off += (size_t)N * 32;
  float* as2    = w + off; off += (size_t)N;
  float* ad2    = w + off; off += (size_t)N;
  float* m2     = w + off; off += (size_t)N;
  float* s2     = w + off; off += (size_t)N;
  float* alpha2 = w + off; off += (size_t)E;
  float* out2   = w + off; off += (size_t)N * 32;
  float* pooled = w + off; off += (size_t)G * 32;
  float* cnt    = w + off; off += (size_t)G;
  float* Me1    = w + off; off += 32;
  float* Me2    = w + off; off += 8;
  float* W1t    = w + off; off += 1024;
  float* W2t    = w + off; off += 4096;
  (void)ws_size; (void)n_in;

  const int TB = 256;
  const float NEG_INF = -__builtin_huge_valf();
  auto blocks = [](int n, int tb) { return (n + tb - 1) / tb; };

  // init accumulators
  fill_f32<<<blocks(N * 4, TB),   TB, 0, stream>>>(m1, NEG_INF, N * 4);
  fill_f32<<<blocks(N * 4, TB),   TB, 0, stream>>>(s1, 0.f,     N * 4);
  fill_f32<<<blocks(N * 128, TB), TB, 0, stream>>>(out1, 0.f,   N * 128);
  fill_f32<<<blocks(N, TB),       TB, 0, stream>>>(m2, NEG_INF, N);
  fill_f32<<<blocks(N, TB),       TB, 0, stream>>>(s2, 0.f,     N);
  fill_f32<<<blocks(N * 32, TB),  TB, 0, stream>>>(out2, 0.f,   N * 32);
  fill_f32<<<blocks(G * 32, TB),  TB, 0, stream>>>(pooled, 0.f, G * 32);
  fill_f32<<<blocks(G, TB),       TB, 0, stream>>>(cnt, 0.f,    G);

  // stage operands: fold att_e, transpose+pad weights, pad x
  fold_atte<<<1, 64, 0, stream>>>(We1, att_e1, We2, att_e2, Me1, Me2);
  prep_weights<<<blocks(1024 + 4096, TB), TB, 0, stream>>>(W1, W2, W1t, W2t);
  pad_x<<<blocks(N * 8, TB), TB, 0, stream>>>(x, xpad, N * 8);

  // -------- layer 1 --------
  gemm_x_w1<<<N / 16, 32, 0, stream>>>(xpad, W1t, h1);
  node_att1<<<blocks(N * 4, TB), TB, 0, stream>>>(h1, att_s1, att_d1, as1, ad1, N);
  edge_alpha1<<<blocks(E, TB), TB, 0, stream>>>(src, dst, eattr, Me1, as1, ad1, alpha1, m1, E);
  edge_exp1<<<blocks(E, TB), TB, 0, stream>>>(dst, m1, alpha1, s1, E);
  edge_msg1<<<blocks(E * 32, TB), TB, 0, stream>>>(src, dst, h1, alpha1, s1, out1, E * 32);

  // bias + BN + ELU (hbn aliases h1 — h1 is dead after edge_msg1)
  float* hbn = h1;
  bn_elu<<<blocks(N * 128, TB), TB, 0, stream>>>(out1, b1, bn_g, bn_b, bn_m, bn_v, hbn, N * 128);

  // -------- layer 2 --------
  gemm_h_w2<<<N / 16, 32, 0, stream>>>(hbn, W2t, h2);
  node_att2<<<blocks(N, TB), TB, 0, stream>>>(h2, att_s2, att_d2, as2, ad2, N);
  edge_alpha2<<<blocks(E, TB), TB, 0, stream>>>(src, dst, eattr, Me2, as2, ad2, alpha2, m2, E);
  edge_exp2<<<blocks(E, TB), TB, 0, stream>>>(dst, m2, alpha2, s2, E);
  edge_msg2<<<blocks(E * 8, TB), TB, 0, stream>>>(src, dst, h2, alpha2, s2, out2, E * 8);

  // -------- ELU + pool + head --------
  elu_pool<<<blocks(N * 32, TB), TB, 0, stream>>>(out2, b2, batch, pooled, cnt, N);
  head<<<blocks(G, TB), TB, 0, stream>>>(pooled, cnt, Wh, bh, out, G);
}